// Encoder1_19628000542732
// MI455X (gfx1250) — compile-verified
//
#include <hip/hip_runtime.h>
#include <hip/hip_bf16.h>

typedef float v2f __attribute__((ext_vector_type(2)));
typedef float v8f __attribute__((ext_vector_type(8)));

#define N_NODES  100000
#define N_EDGES  3200000
#define N_GRAPHS 128
#define DF       256
#define H_MLP    512
#define N_LAYERS 3
#define BN_EPS   1e-5f
#define GEMM_MT  5              /* 16*5 = 80 rows per wave; 100000 = 80*1250 */

#if defined(__has_builtin)
#if __has_builtin(__builtin_amdgcn_sched_group_barrier)
#define SCHED_GROUP(mask, size, id) __builtin_amdgcn_sched_group_barrier(mask, size, id)
#endif
#endif
#ifndef SCHED_GROUP
#define SCHED_GROUP(mask, size, id)
#endif

// ---------------------------------------------------------------- utilities
__global__ void zero_kernel(float* __restrict__ p, long long n) {
    long long i = (long long)blockIdx.x * blockDim.x + threadIdx.x;
    if (i < n) p[i] = 0.0f;
}

// ------------------------------------------------------------- CSR building
__global__ void degree_kernel(const int* __restrict__ dst, int* __restrict__ deg,
                              int nEdges) {
    int e = blockIdx.x * blockDim.x + threadIdx.x;
    if (e < nEdges) atomicAdd(&deg[dst[e]], 1);
}

// block-level exclusive scan (1024 elems/block), emits per-block totals
__global__ void scan_block_kernel(const int* __restrict__ deg, int* __restrict__ ex,
                                  int* __restrict__ blockSums, int n) {
    __shared__ int tmp[1024];
    int t = threadIdx.x;
    int g = blockIdx.x * 1024 + t;
    int v = (g < n) ? deg[g] : 0;
    tmp[t] = v;
    __syncthreads();
    for (int off = 1; off < 1024; off <<= 1) {
        int add = (t >= off) ? tmp[t - off] : 0;
        __syncthreads();
        tmp[t] += add;
        __syncthreads();
    }
    if (g < n) ex[g] = tmp[t] - v;                 // exclusive value
    if (t == 1023) blockSums[blockIdx.x] = tmp[t]; // inclusive block total
}

// single-block exclusive scan of the block totals (<=128 entries)
__global__ void scan_top_kernel(int* __restrict__ blockSums, int nb) {
    __shared__ int tmp[128];
    int t = threadIdx.x;
    int v = (t < nb) ? blockSums[t] : 0;
    tmp[t] = v;
    __syncthreads();
    for (int off = 1; off < 128; off <<= 1) {
        int add = (t >= off) ? tmp[t - off] : 0;
        __syncthreads();
        tmp[t] += add;
        __syncthreads();
    }
    if (t < nb) blockSums[t] = tmp[t] - v;
}

// add block offsets; also initialize the fill cursor
__global__ void scan_add_kernel(int* __restrict__ ex, const int* __restrict__ blockSums,
                                int* __restrict__ cursor, int n) {
    int g = blockIdx.x * 1024 + threadIdx.x;
    if (g < n) {
        int v = ex[g] + blockSums[blockIdx.x];
        ex[g] = v;
        cursor[g] = v;
    }
}

__global__ void csr_fill_kernel(const int* __restrict__ src, const int* __restrict__ dst,
                                int* __restrict__ cursor, int* __restrict__ csrSrc,
                                int nEdges) {
    int e = blockIdx.x * blockDim.x + threadIdx.x;
    if (e < nEdges) {
        int pos = atomicAdd(&cursor[dst[e]], 1);
        csrSrc[pos] = src[e];
    }
}

// -------------------------------------------------------------- aggregation
// x[n] = h[n] + sum_{s in in-neighbors(n)} h[s]
// One wave per node; each lane owns two float4 feature chunks (256 floats).
// Pure gather from the L2-resident node table: zero atomics.
__global__ void gather_agg_kernel(const float* __restrict__ h,
                                  const int* __restrict__ rowStart,
                                  const int* __restrict__ csrSrc,
                                  float* __restrict__ x,
                                  int nNodes, int nEdges) {
    int wave = (int)(((long long)blockIdx.x * blockDim.x + threadIdx.x) >> 5);
    int lane = threadIdx.x & 31;
    if (wave >= nNodes) return;
    const float4* hv = (const float4*)h;
    size_t rbase = (size_t)wave * 64;
    float4 a0 = hv[rbase + lane];
    float4 a1 = hv[rbase + lane + 32];
    int i  = rowStart[wave];
    int i1 = (wave + 1 < nNodes) ? rowStart[wave + 1] : nEdges;
    for (; i < i1; ++i) {
        int s = csrSrc[i];
        size_t sb = (size_t)s * 64;
        float4 b0 = hv[sb + lane];
        float4 b1 = hv[sb + lane + 32];
        a0.x += b0.x; a0.y += b0.y; a0.z += b0.z; a0.w += b0.w;
        a1.x += b1.x; a1.y += b1.y; a1.z += b1.z; a1.w += b1.w;
    }
    ((float4*)x)[rbase + lane]      = a0;
    ((float4*)x)[rbase + lane + 32] = a1;
}

// ------------------------------------------------------------------- GEMM
// C[N x NC] = A[N x K] * B[K x NC] via V_WMMA_F32_16X16X4_F32.
// Workgroup = 8 waves sharing one 16-col N-tile; B tile (K x 16) staged in
// LDS in K-paired layout -> fragment = one conflict-free ds_load_b64.
// Each wave computes GEMM_MT stacked 16x16 tiles (one B frag -> 5 WMMAs).
// K-loop is software-pipelined depth-2 (two K-steps of fragments in flight)
// and sched_group_barrier forces {1 DS-read, 5 VMEM-reads} -> {5 WMMA}
// interleave so WMMAs wait on loads issued two steps earlier.
__global__ __launch_bounds__(256)
void gemm_f32_wmma(const float* __restrict__ A, const float* __restrict__ B,
                   float* __restrict__ C, int N, int K, int NC, int mgroups) {
    __shared__ float bTile[H_MLP * 16];            // (K/2)*32 floats, K<=512 -> 32KB

    int ntiles = NC >> 4;
    int bm     = blockIdx.x / ntiles;
    int ntile  = blockIdx.x - bm * ntiles;
    int n0     = ntile * 16;

    // cooperative stage of B tile: element (kk,n) -> lds[(kk>>1)*32 + n*2 + (kk&1)]
    {
        int n = threadIdx.x & 15;
        for (int kk = threadIdx.x >> 4; kk < K; kk += 16)
            bTile[(kk >> 1) * 32 + n * 2 + (kk & 1)] = B[(size_t)kk * NC + n0 + n];
    }
    __syncthreads();

    int wave   = threadIdx.x >> 5;
    int lane   = threadIdx.x & 31;
    int mgroup = bm * 8 + wave;
    if (mgroup >= mgroups) return;                 // wave-uniform, after barrier

    int m0    = mgroup * (16 * GEMM_MT);
    int lrow  = lane & 15;
    int khalf = (lane >> 4) * 2;                   // 0 or 2

    v8f acc[GEMM_MT];
#pragma unroll
    for (int i = 0; i < GEMM_MT; ++i) acc[i] = (v8f){0,0,0,0,0,0,0,0};

    const float* ap[GEMM_MT];
#pragma unroll
    for (int i = 0; i < GEMM_MT; ++i)
        ap[i] = A + (size_t)(m0 + i * 16 + lrow) * K + khalf;
    const float* bp = bTile + khalf * 16 + lrow * 2;   // + k*16 per step

    // ---- software pipeline prologue: stages for k=0 and k=4 ----
    v2f b0 = *(const v2f*)(bp);
    v2f b1 = *(const v2f*)(bp + 4 * 16);
    v2f a0[GEMM_MT], a1[GEMM_MT];
#pragma unroll
    for (int i = 0; i < GEMM_MT; ++i) a0[i] = *(const v2f*)(ap[i]);
#pragma unroll
    for (int i = 0; i < GEMM_MT; ++i) a1[i] = *(const v2f*)(ap[i] + 4);

#pragma unroll 4
    for (int k = 0; k < K - 8; k += 4) {
        // prefetch step k+8 while step k computes (depth 2)
        v2f b2 = *(const v2f*)(bp + (k + 8) * 16);         // ds_load_b64
        v2f a2[GEMM_MT];
#pragma unroll
        for (int i = 0; i < GEMM_MT; ++i)
            a2[i] = *(const v2f*)(ap[i] + k + 8);          // global_load_b64
#pragma unroll
        for (int i = 0; i < GEMM_MT; ++i)
            acc[i] = __builtin_amdgcn_wmma_f32_16x16x4_f32(
                false, a0[i], false, b0, (short)0, acc[i], false, false);
        // rotate stages (SSA copies, renamed by unroller)
        b0 = b1; b1 = b2;
#pragma unroll
        for (int i = 0; i < GEMM_MT; ++i) { a0[i] = a1[i]; a1[i] = a2[i]; }
        // desired issue pattern per step: DS read, 5 VMEM reads, 5 WMMA
        SCHED_GROUP(0x0100, 1, 0);   // DS read
        SCHED_GROUP(0x0020, 5, 0);   // VMEM reads
        SCHED_GROUP(0x0008, 5, 0);   // MFMA/WMMA
    }
    // epilogue: last two K-steps
#pragma unroll
    for (int i = 0; i < GEMM_MT; ++i)
        acc[i] = __builtin_amdgcn_wmma_f32_16x16x4_f32(
            false, a0[i], false, b0, (short)0, acc[i], false, false);
#pragma unroll
    for (int i = 0; i < GEMM_MT; ++i)
        acc[i] = __builtin_amdgcn_wmma_f32_16x16x4_f32(
            false, a1[i], false, b1, (short)0, acc[i], false, false);

#pragma unroll
    for (int i = 0; i < GEMM_MT; ++i) {
        int mbase = m0 + i * 16 + (lane >> 4) * 8;
        float* cp = C + (size_t)mbase * NC + n0 + lrow;
#pragma unroll
        for (int r = 0; r < 8; ++r) cp[(size_t)r * NC] = acc[i][r];
    }
}

// -------------------------------------------------------------- batch-norm
__global__ void colstats_kernel(const float* __restrict__ X, int nRows, int nCols,
                                float* __restrict__ sum, float* __restrict__ sumsq,
                                int rowsPerBlock) {
    int c  = threadIdx.x;                       // blockDim.x == nCols
    int r0 = blockIdx.x * rowsPerBlock;
    int r1 = r0 + rowsPerBlock; if (r1 > nRows) r1 = nRows;
    if (r0 >= nRows) return;
    float s = 0.f, sq = 0.f;
    for (int r = r0; r < r1; ++r) {
        float v = X[(size_t)r * nCols + c];
        s += v; sq += v * v;
    }
    atomicAdd(&sum[c], s);
    atomicAdd(&sumsq[c], sq);
}

__global__ void bn_finalize_kernel(const float* __restrict__ sum,
                                   const float* __restrict__ sumsq,
                                   const float* __restrict__ gamma,
                                   const float* __restrict__ beta,
                                   float* __restrict__ scale,
                                   float* __restrict__ bias,
                                   int nCols, float invN) {
    int c = blockIdx.x * blockDim.x + threadIdx.x;
    if (c >= nCols) return;
    float m   = sum[c] * invN;
    float var = sumsq[c] * invN - m * m;
    float sc  = gamma[c] * rsqrtf(var + BN_EPS);
    scale[c] = sc;
    bias[c]  = beta[c] - m * sc;
}

__global__ void bn_relu_kernel(float* __restrict__ X,
                               const float* __restrict__ scale,
                               const float* __restrict__ bias,
                               long long n4, int colMask4) {
    long long i = (long long)blockIdx.x * blockDim.x + threadIdx.x;
    if (i >= n4) return;
    float4 v = ((float4*)X)[i];
    int c = ((int)i & colMask4) * 4;
    v.x = v.x * scale[c]     + bias[c];     if (v.x < 0.f) v.x = 0.f;
    v.y = v.y * scale[c + 1] + bias[c + 1]; if (v.y < 0.f) v.y = 0.f;
    v.z = v.z * scale[c + 2] + bias[c + 2]; if (v.z < 0.f) v.z = 0.f;
    v.w = v.w * scale[c + 3] + bias[c + 3]; if (v.w < 0.f) v.w = 0.f;
    ((float4*)X)[i] = v;
}

// ------------------------------------------------------------------ pooling
__global__ void pool_kernel(const float* __restrict__ h,
                            const int* __restrict__ gid,
                            float* __restrict__ pooled,   // [N_GRAPHS x 3*DF]
                            int nRows, int rowsPerBlock, int colOffset) {
    int c  = threadIdx.x;                        // blockDim.x == DF
    int r0 = blockIdx.x * rowsPerBlock;
    int r1 = r0 + rowsPerBlock; if (r1 > nRows) r1 = nRows;
    if (r0 >= nRows) return;
    float acc = 0.f;
    int g = gid[r0];
    for (int r = r0; r < r1; ++r) {
        int gr = gid[r];
        if (gr != g) {
            atomicAdd(&pooled[(size_t)g * (N_LAYERS * DF) + colOffset + c], acc);
            acc = 0.f; g = gr;
        }
        acc += h[(size_t)r * DF + c];
    }
    atomicAdd(&pooled[(size_t)g * (N_LAYERS * DF) + colOffset + c], acc);
}

// ------------------------------------------------------------------ driver
extern "C" void kernel_launch(void* const* d_in, const int* in_sizes, int n_in,
                              void* d_out, int out_size, void* d_ws, size_t ws_size,
                              hipStream_t stream) {
    const float* h_in      = (const float*)d_in[0];
    const float* W1        = (const float*)d_in[1];   // [L,256,512]
    const float* W2        = (const float*)d_in[2];   // [L,512,256]
    const float* bn1_gamma = (const float*)d_in[3];
    const float* bn1_beta  = (const float*)d_in[4];
    const float* bn2_gamma = (const float*)d_in[5];
    const float* bn2_beta  = (const float*)d_in[6];
    const int*   src       = (const int*)d_in[7];
    const int*   dst       = (const int*)d_in[8];
    const int*   graph_id  = (const int*)d_in[9];

    const long long nH = (long long)N_NODES * DF;     // 25,600,000
    const long long nT = (long long)N_NODES * H_MLP;  // 51,200,000

    float* bufX  = (float*)d_ws;                      // [N x 256]
    float* bufT  = bufX + nH;                         // [N x 512]
    float* sum   = bufT + nT;
    float* sumsq = sum + H_MLP;
    float* scale = sumsq + H_MLP;
    float* bias  = scale + H_MLP;
    int*   deg       = (int*)(bias + H_MLP);          // [N]
    int*   rowStart  = deg + N_NODES;                 // [N]
    int*   cursor    = rowStart + N_NODES;            // [N]
    int*   blockSums = cursor + N_NODES;              // [128]
    int*   csrSrc    = blockSums + 128;               // [E]

    float* outH   = (float*)d_out;                    // [N x 256] final h
    float* pooled = outH + nH;                        // [128 x 768]

    const int TB = 256;
    const int scanBlocks = (N_NODES + 1023) / 1024;   // 98

    // zero pooled output + degree histogram
    {
        long long n = (long long)N_GRAPHS * N_LAYERS * DF;
        zero_kernel<<<(int)((n + TB - 1) / TB), TB, 0, stream>>>(pooled, n);
        zero_kernel<<<(N_NODES + TB - 1) / TB, TB, 0, stream>>>((float*)deg, N_NODES);
    }

    // ---- build CSR once per call (graph is static across layers) ----
    degree_kernel<<<(N_EDGES + TB - 1) / TB, TB, 0, stream>>>(dst, deg, N_EDGES);
    scan_block_kernel<<<scanBlocks, 1024, 0, stream>>>(deg, rowStart, blockSums, N_NODES);
    scan_top_kernel<<<1, 128, 0, stream>>>(blockSums, scanBlocks);
    scan_add_kernel<<<scanBlocks, 1024, 0, stream>>>(rowStart, blockSums, cursor, N_NODES);
    csr_fill_kernel<<<(N_EDGES + TB - 1) / TB, TB, 0, stream>>>(src, dst, cursor, csrSrc, N_EDGES);

    const int mgroups = N_NODES / (16 * GEMM_MT);     // 1250
    const int blocksM = (mgroups + 7) / 8;            // 157

    const float* hcur = h_in;
    for (int l = 0; l < N_LAYERS; ++l) {
        // x = h + segment_sum(h[src], dst) : pure CSR gather, no atomics
        {
            long long nt = (long long)N_NODES * 32;
            gather_agg_kernel<<<(int)((nt + TB - 1) / TB), TB, 0, stream>>>(
                hcur, rowStart, csrSrc, bufX, N_NODES, N_EDGES);
        }

        // t = x @ W1[l]    [N x 256] * [256 x 512]
        gemm_f32_wmma<<<blocksM * (H_MLP / 16), TB, 0, stream>>>(
            bufX, W1 + (size_t)l * DF * H_MLP, bufT, N_NODES, DF, H_MLP, mgroups);

        // inner BN + ReLU (512 cols)
        {
            zero_kernel<<<(2 * H_MLP + TB - 1) / TB, TB, 0, stream>>>(sum, 2 * H_MLP);
            colstats_kernel<<<(N_NODES + 255) / 256, H_MLP, 0, stream>>>(
                bufT, N_NODES, H_MLP, sum, sumsq, 256);
            bn_finalize_kernel<<<1, H_MLP, 0, stream>>>(
                sum, sumsq, bn1_gamma + (size_t)l * H_MLP, bn1_beta + (size_t)l * H_MLP,
                scale, bias, H_MLP, 1.0f / N_NODES);
            bn_relu_kernel<<<(int)((nT / 4 + TB - 1) / TB), TB, 0, stream>>>(
                bufT, scale, bias, nT / 4, (H_MLP >> 2) - 1);
        }

        // h = t @ W2[l]    [N x 512] * [512 x 256] -> outH
        gemm_f32_wmma<<<blocksM * (DF / 16), TB, 0, stream>>>(
            bufT, W2 + (size_t)l * H_MLP * DF, outH, N_NODES, H_MLP, DF, mgroups);

        // outer BN + ReLU (256 cols)
        {
            zero_kernel<<<(2 * DF + TB - 1) / TB, TB, 0, stream>>>(sum, 2 * DF);
            colstats_kernel<<<(N_NODES + 255) / 256, DF, 0, stream>>>(
                outH, N_NODES, DF, sum, sumsq, 256);
            bn_finalize_kernel<<<1, DF, 0, stream>>>(
                sum, sumsq, bn2_gamma + (size_t)l * DF, bn2_beta + (size_t)l * DF,
                scale, bias, DF, 1.0f / N_NODES);
            bn_relu_kernel<<<(int)((nH / 4 + TB - 1) / TB), TB, 0, stream>>>(
                outH, scale, bias, nH / 4, (DF >> 2) - 1);
        }

        // sum-pool into pooled[:, l*256:(l+1)*256]
        pool_kernel<<<(N_NODES + 255) / 256, DF, 0, stream>>>(
            outH, graph_id, pooled, N_NODES, 256, l * DF);

        hcur = outH;
    }
}